// SuperGATEncoder_9620726743404
// MI455X (gfx1250) — compile-verified
//
#include <hip/hip_runtime.h>
#include <hip/hip_bf16.h>

typedef float v2f __attribute__((ext_vector_type(2)));
typedef float v8f __attribute__((ext_vector_type(8)));

#define NEG_SLOPE 0.2f
#define HEADS 2

// ---------------------------------------------------------------------------
// float -> order-preserving unsigned mapping (for atomicMax-based segment max)
// ---------------------------------------------------------------------------
__device__ __forceinline__ unsigned fmap(float f) {
  unsigned u = __float_as_uint(f);
  return (u & 0x80000000u) ? ~u : (u | 0x80000000u);
}
__device__ __forceinline__ float funmap(unsigned u) {
  unsigned v = (u & 0x80000000u) ? (u & 0x7FFFFFFFu) : ~u;
  return __uint_as_float(v);
}

// ---------------------------------------------------------------------------
// WMMA GEMM: C[M,NC] = A[M,K] @ B[K,NC], K/NC compile-time.
// Block = 256 threads (8 waves). Whole B staged in LDS, permuted into
// V_WMMA_F32_16X16X4_F32 fragment order so each B-frag is one ds_load_b64.
// Each wave computes a 16-row x NC strip: A-frag (one global_load_b64 per
// K-step, row clamped instead of predicated) is reused across all NC/16 tiles.
// ---------------------------------------------------------------------------
template <int K, int NC>
__global__ __launch_bounds__(256) void gemm_wmma_kernel(
    const float* __restrict__ A, const float* __restrict__ B,
    float* __restrict__ C, int M) {
  constexpr int NT    = NC / 16;  // N tiles per strip
  constexpr int KSTEP = K / 4;    // wmma steps
  // Bs layout: [(k>>2)*NC + n]*4 + hf*2 + (k&1), hf = (k>>1)&1
  __shared__ float Bs[K * NC];

  const int tid = threadIdx.x;
  // ---- stage B (coalesced read, permuted LDS write) ----
  for (int idx = tid; idx < K * NC; idx += 256) {
    const int k = idx / NC;  // NC is constexpr -> shifts
    const int n = idx % NC;
    Bs[((k >> 2) * NC + n) * 4 + ((k >> 1) & 1) * 2 + (k & 1)] = B[idx];
  }
  __syncthreads();

  const int wave = tid >> 5;
  const int lane = tid & 31;
  const int hf   = lane >> 4;  // lane half
  const int lr   = lane & 15;
  const int m0   = blockIdx.x * 128 + wave * 16;

  // clamp row for loads; out-of-range rows only feed D-rows masked at store
  int rowA = m0 + lr;
  if (rowA > M - 1) rowA = M - 1;
  const float* Arow = A + (long long)rowA * K + 2 * hf;

  v8f acc[NT];
#pragma unroll
  for (int t = 0; t < NT; ++t) acc[t] = {};

#pragma unroll
  for (int ks = 0; ks < KSTEP; ++ks) {
    const v2f a = *(const v2f*)(Arow + ks * 4);  // one global_load_b64
#pragma unroll
    for (int t = 0; t < NT; ++t) {
      const v2f b = *(const v2f*)(&Bs[(ks * NC + t * 16 + lr) * 4 + hf * 2]);
      acc[t] = __builtin_amdgcn_wmma_f32_16x16x4_f32(
          false, a, false, b, (short)0, acc[t], false, false);
    }
  }

  // C/D 16x16 f32 layout: VGPR i -> M = i (lanes 0-15) / M = 8+i (lanes 16-31)
  const int rowBase = m0 + (hf << 3);
#pragma unroll
  for (int t = 0; t < NT; ++t) {
    const int col = t * 16 + lr;
#pragma unroll
    for (int i = 0; i < 8; ++i) {
      const int r = rowBase + i;
      if (r < M) C[(long long)r * NC + col] = acc[t][i];
    }
  }
}

// ---------------------------------------------------------------------------
// Fill kernels (harness poisons d_out/d_ws -> must zero-init every call)
// ---------------------------------------------------------------------------
__global__ void fill_f32(float* __restrict__ p, float v, long long n) {
  long long i = (long long)blockIdx.x * blockDim.x + threadIdx.x;
  if (i < n) p[i] = v;
}
__global__ void fill_u32(unsigned* __restrict__ p, unsigned v, long long n) {
  long long i = (long long)blockIdx.x * blockDim.x + threadIdx.x;
  if (i < n) p[i] = v;
}

// ---------------------------------------------------------------------------
// Edges 0..E-1 come from edge_index, edges E..Etot-1 are self loops
// ---------------------------------------------------------------------------
__device__ __forceinline__ void edge_sd(const int* __restrict__ ei, int Ereal,
                                        long long e, int& s, int& d) {
  if (e < Ereal) { s = ei[e]; d = ei[(long long)Ereal + e]; }
  else           { s = d = (int)(e - Ereal); }
}

// alpha = leaky_relu( (<hj,att_l> + <hi,att_r>) * sigmoid(<hi,hj>) ), + seg max
template <int C>
__global__ void edge_alpha_kernel(
    const float* __restrict__ G, const int* __restrict__ ei, int Ereal, int Etot,
    const float* __restrict__ att_l, const float* __restrict__ att_r,
    float* __restrict__ alpha, unsigned* __restrict__ amaxU) {
  long long t = (long long)blockIdx.x * blockDim.x + threadIdx.x;
  long long nwork = (long long)Etot * HEADS;
  if (t >= nwork) return;
  long long e = t >> 1;
  int h = (int)(t & 1);
  int s, d; edge_sd(ei, Ereal, e, s, d);

  const float* gj = G + ((long long)s * HEADS + h) * C;  // source (x_j)
  const float* gi = G + ((long long)d * HEADS + h) * C;  // target (x_i)
  const float* al = att_l + h * C;
  const float* ar = att_r + h * C;

  float logits = 0.0f, a = 0.0f;
#pragma unroll 4
  for (int c = 0; c < C; c += 4) {
    float4 vj = *(const float4*)(gj + c);
    float4 vi = *(const float4*)(gi + c);
    float4 vl = *(const float4*)(al + c);
    float4 vr = *(const float4*)(ar + c);
    logits += vj.x * vi.x + vj.y * vi.y + vj.z * vi.z + vj.w * vi.w;
    a      += vj.x * vl.x + vj.y * vl.y + vj.z * vl.z + vj.w * vl.w;
    a      += vi.x * vr.x + vi.y * vr.y + vi.z * vr.z + vi.w * vr.w;
  }
  float sig = 1.0f / (1.0f + __expf(-logits));
  a *= sig;
  a = (a > 0.0f) ? a : NEG_SLOPE * a;  // leaky relu
  alpha[t] = a;
  atomicMax(&amaxU[(long long)d * HEADS + h], fmap(a));
}

// ea = exp(alpha - segmax[dst]); denom[dst] += ea  (ea overwrites alpha)
__global__ void edge_softmax_kernel(
    const int* __restrict__ ei, int Ereal, int Etot,
    float* __restrict__ alpha, const unsigned* __restrict__ amaxU,
    float* __restrict__ denom) {
  long long t = (long long)blockIdx.x * blockDim.x + threadIdx.x;
  long long nwork = (long long)Etot * HEADS;
  if (t >= nwork) return;
  long long e = t >> 1;
  int h = (int)(t & 1);
  int s, d; edge_sd(ei, Ereal, e, s, d);
  float m  = funmap(amaxU[(long long)d * HEADS + h]);
  float ea = __expf(alpha[t] - m);
  alpha[t] = ea;
  atomicAdd(&denom[(long long)d * HEADS + h], ea);
}

// out[dst,h,c4] += G[src,h,c4] * ea/(denom+eps), one thread per (edge,h,4 chans)
template <int C>
__global__ void edge_aggregate_kernel(
    const float* __restrict__ G, const int* __restrict__ ei, int Ereal, int Etot,
    const float* __restrict__ ea, const float* __restrict__ denom,
    float* __restrict__ out) {
  constexpr int CQ = C / 4;
  long long t = (long long)blockIdx.x * blockDim.x + threadIdx.x;
  long long nwork = (long long)Etot * HEADS * CQ;
  if (t >= nwork) return;
  int c4 = (int)(t % CQ);
  int h  = (int)((t / CQ) & 1);
  long long e = t / (CQ * HEADS);
  int s, d; edge_sd(ei, Ereal, e, s, d);
  float w = ea[e * HEADS + h] / (denom[(long long)d * HEADS + h] + 1e-16f);
  float4 v = *(const float4*)(G + ((long long)s * HEADS + h) * C + c4 * 4);
  float* o = out + ((long long)d * HEADS + h) * C + c4 * 4;
  atomicAdd(o + 0, v.x * w);
  atomicAdd(o + 1, v.y * w);
  atomicAdd(o + 2, v.z * w);
  atomicAdd(o + 3, v.w * w);
}

// y += bias (broadcast over rows), optional ELU
__global__ void bias_act_kernel(float* __restrict__ y, const float* __restrict__ b,
                                long long n, int F, int do_elu) {
  long long i = (long long)blockIdx.x * blockDim.x + threadIdx.x;
  if (i >= n) return;
  float v = y[i] + b[(int)(i % F)];
  if (do_elu) v = (v > 0.0f) ? v : expm1f(v);
  y[i] = v;
}

// ---------------------------------------------------------------------------
// Host-side orchestration
// ---------------------------------------------------------------------------
static inline int cdivi(long long a, long long b) { return (int)((a + b - 1) / b); }

template <int C>
static void run_layer(const float* Xin, const float* W,
                      const float* attL, const float* attR, const float* bias,
                      float* G, float* out,
                      float* alpha, unsigned* amaxU, float* denom,
                      const int* ei, int Ereal, int Etot, int Nn, int do_elu,
                      hipStream_t stream) {
  constexpr int F = HEADS * C;
  // GEMM: G = Xin @ W   [Nn,128] x [128,F]
  gemm_wmma_kernel<128, F><<<cdivi(Nn, 128), 256, 0, stream>>>(Xin, W, G, Nn);
  // init segment-max / denom / output accumulator
  fill_u32<<<cdivi((long long)Nn * HEADS, 256), 256, 0, stream>>>(
      amaxU, 0u, (long long)Nn * HEADS);
  fill_f32<<<cdivi((long long)Nn * HEADS, 256), 256, 0, stream>>>(
      denom, 0.0f, (long long)Nn * HEADS);
  fill_f32<<<cdivi((long long)Nn * F, 256), 256, 0, stream>>>(
      out, 0.0f, (long long)Nn * F);
  // attention
  long long ew = (long long)Etot * HEADS;
  edge_alpha_kernel<C><<<cdivi(ew, 256), 256, 0, stream>>>(
      G, ei, Ereal, Etot, attL, attR, alpha, amaxU);
  edge_softmax_kernel<<<cdivi(ew, 256), 256, 0, stream>>>(
      ei, Ereal, Etot, alpha, amaxU, denom);
  long long aw = (long long)Etot * HEADS * (C / 4);
  edge_aggregate_kernel<C><<<cdivi(aw, 256), 256, 0, stream>>>(
      G, ei, Ereal, Etot, alpha, denom, out);
  bias_act_kernel<<<cdivi((long long)Nn * F, 256), 256, 0, stream>>>(
      out, bias, (long long)Nn * F, F, do_elu);
}

extern "C" void kernel_launch(void* const* d_in, const int* in_sizes, int n_in,
                              void* d_out, int out_size, void* d_ws, size_t ws_size,
                              hipStream_t stream) {
  const float* x      = (const float*)d_in[0];
  const int*   ei     = (const int*)  d_in[1];
  const float* W1     = (const float*)d_in[2];
  const float* attL1  = (const float*)d_in[3];
  const float* attR1  = (const float*)d_in[4];
  const float* b1     = (const float*)d_in[5];
  const float* Wmu    = (const float*)d_in[6];
  const float* attLmu = (const float*)d_in[7];
  const float* attRmu = (const float*)d_in[8];
  const float* bmu    = (const float*)d_in[9];
  const float* Wls    = (const float*)d_in[10];
  const float* attLls = (const float*)d_in[11];
  const float* attRls = (const float*)d_in[12];
  const float* bls    = (const float*)d_in[13];

  const int IN = 128, C1v = 64, C2v = 32;
  const int F1 = HEADS * C1v;           // 128
  const int F2 = HEADS * C2v;           // 64
  const int Nn = in_sizes[0] / IN;      // 50000
  const int Er = in_sizes[1] / 2;       // 800000
  const int Et = Er + Nn;               // + self loops

  // workspace layout (floats)
  float*    G     = (float*)d_ws;                      // Nn * F1 (reused per layer)
  float*    Hbuf  = G    + (long long)Nn * F1;         // Nn * F1
  float*    alpha = Hbuf + (long long)Nn * F1;         // Et * HEADS
  unsigned* amaxU = (unsigned*)(alpha + (long long)Et * HEADS);  // Nn * HEADS
  float*    denom = (float*)(amaxU + (long long)Nn * HEADS);     // Nn * HEADS

  float* mu = (float*)d_out;
  float* ls = mu + (long long)Nn * F2;

  // layer 1: h = ELU(agg(x @ W1) + b1)
  run_layer<C1v>(x, W1, attL1, attR1, b1, G, Hbuf,
                 alpha, amaxU, denom, ei, Er, Et, Nn, /*elu*/1, stream);
  // mu head
  run_layer<C2v>(Hbuf, Wmu, attLmu, attRmu, bmu, G, mu,
                 alpha, amaxU, denom, ei, Er, Et, Nn, /*elu*/0, stream);
  // logstd head
  run_layer<C2v>(Hbuf, Wls, attLls, attRls, bls, G, ls,
                 alpha, amaxU, denom, ei, Er, Et, Nn, /*elu*/0, stream);
}